// AtomEmbedding_78658031059448
// MI455X (gfx1250) — compile-verified
//
#include <hip/hip_runtime.h>
#include <hip/hip_bf16.h>

// ---------------- problem constants (match reference) ----------------
#define NBX   4
#define LRES  256
#define AAT   15
#define PAT   (LRES*AAT)        // 3840 atoms per batch
#define NATOT (NBX*PAT)         // 15360 atoms total
#define KNN   16
#define NEDGE (NATOT*KNN)       // 245760 edges
#define FEATD 128
#define NRBF  32
#define CUT   5.0f

typedef __attribute__((ext_vector_type(2))) float v2f;
typedef __attribute__((ext_vector_type(8))) float v8f;

// ---------------- kernel 1: per-atom class index "aaa" ----------------
__global__ void k_aaa(const int* __restrict__ mask_atoms,
                      const int* __restrict__ seq_mask,
                      int* __restrict__ aaa) {
  int i = blockIdx.x * blockDim.x + threadIdx.x;
  if (i >= NATOT) return;
  int atom = i % AAT;
  int res  = (i / AAT) % LRES;
  int b    = i / PAT;
  int v = mask_atoms[i] ? atom : 15;                 // MAX_AAA-1 = 15
  if (!(seq_mask[b * LRES + res] || v == 15)) v = 16; // MAX_AAA
  aaa[i] = v;
}

// ---------------- kernel 2: brute-force kNN (per batch, LDS tiles) ----
// One thread = one query atom; top-16 kept sorted (ascending d2, ties by
// ascending index, matching jax.lax.top_k of -d2) in registers.
__global__ __launch_bounds__(256) void k_knn(const float* __restrict__ pos,
                                             const int*   __restrict__ mask,
                                             int* __restrict__ srcIdx,
                                             int* __restrict__ emaskE,
                                             int* __restrict__ edge_index_out) {
  __shared__ float4 tile[256];
  const int b  = blockIdx.y;
  const int q  = blockIdx.x * 256 + threadIdx.x;   // 0..PAT-1
  const int qg = b * PAT + q;
  const float qx = pos[qg * 3 + 0];
  const float qy = pos[qg * 3 + 1];
  const float qz = pos[qg * 3 + 2];
  const int qvalid = mask[qg];

  float bd[KNN]; int bi[KNN];
#pragma unroll
  for (int k = 0; k < KNN; ++k) { bd[k] = __builtin_inff(); bi[k] = 0; }

  for (int t = 0; t < PAT; t += 256) {
    int cg = b * PAT + t + threadIdx.x;
    float4 c;
    c.x = pos[cg * 3 + 0]; c.y = pos[cg * 3 + 1]; c.z = pos[cg * 3 + 2];
    c.w = mask[cg] ? 1.0f : 0.0f;
    __syncthreads();
    tile[threadIdx.x] = c;
    __syncthreads();
    if (qvalid) {
      for (int j = 0; j < 256; ++j) {
        float4 p = tile[j];
        float dx = qx - p.x, dy = qy - p.y, dz = qz - p.z;
        float d2 = dx * dx + dy * dy + dz * dz;
        if (p.w != 0.0f && d2 < CUT * CUT && d2 < bd[KNN - 1]) {
          int jj = t + j;
          // branchless sorted insertion (strict compares keep earlier-index
          // equal entries ahead -> matches top_k tie-break)
#pragma unroll
          for (int k = KNN - 1; k > 0; --k) {
            bool shift = d2 < bd[k - 1];
            bool place = (d2 < bd[k]) && !shift;
            float nv = shift ? bd[k - 1] : (place ? d2 : bd[k]);
            int   ni = shift ? bi[k - 1] : (place ? jj : bi[k]);
            bd[k] = nv; bi[k] = ni;
          }
          if (d2 < bd[0]) { bd[0] = d2; bi[0] = jj; }
        }
      }
    }
  }
#pragma unroll
  for (int k = 0; k < KNN; ++k) {
    int e  = qg * KNN + k;
    int em = (bd[k] < 1e30f) ? 1 : 0;
    int sg = em ? (b * PAT + bi[k]) : qg;
    srcIdx[e] = sg;
    emaskE[e] = em;
    edge_index_out[e]         = sg;   // src row of edge_index
    edge_index_out[NEDGE + e] = qg;   // dst row of edge_index
  }
}

// ---------------- kernel 3: per-edge geometry / RBF / sh / keep -------
__global__ void k_edge(const float* __restrict__ pos,
                       const int*   __restrict__ srcIdx,
                       const int*   __restrict__ emaskE,
                       const int*   __restrict__ smask,
                       float* __restrict__ edge_attr,   // E x 32
                       float* __restrict__ scaleW,      // E   (C * mmask)
                       float* __restrict__ out_ew,      // E
                       float* __restrict__ out_sh,      // E x 8
                       float* __restrict__ out_keep) {  // E
  int e = blockIdx.x * blockDim.x + threadIdx.x;
  if (e >= NEDGE) return;
  int sg = srcIdx[e];
  int dg = e / KNN;
  int emi = emaskE[e];
  float em = emi ? 1.0f : 0.0f;
  float rx = (pos[sg * 3 + 0] - pos[dg * 3 + 0]) * em;
  float ry = (pos[sg * 3 + 1] - pos[dg * 3 + 1]) * em;
  float rz = (pos[sg * 3 + 2] - pos[dg * 3 + 2]) * em;
  float w2 = rx * rx + ry * ry + rz * rz;
  float ew = (w2 > 0.0f) ? sqrtf(w2) : 0.0f;

  const float PI_F = 3.14159265358979323846f;
  float C = 0.5f * (__cosf(ew * PI_F / CUT) + 1.0f) * ((ew < CUT) ? 1.0f : 0.0f);

  // expnorm RBF: alpha=1, start=exp(-5), beta=(2/32*(1-start))^-2
  const float start = 0.006737946999085467f;
  const float tmp   = 0.0625f * (1.0f - start);
  const float beta  = 1.0f / (tmp * tmp);
  const float step  = (1.0f - start) / (NRBF - 1);
  float ed = __expf(-ew);
  // compute 32 RBFs into registers, store as 8 x b128
  float4* eav = (float4*)(edge_attr + (size_t)e * NRBF);
#pragma unroll
  for (int r4 = 0; r4 < NRBF / 4; ++r4) {
    float4 v;
    {
      float m0 = start + step * (float)(4 * r4 + 0); float t0 = ed - m0;
      float m1 = start + step * (float)(4 * r4 + 1); float t1 = ed - m1;
      float m2 = start + step * (float)(4 * r4 + 2); float t2 = ed - m2;
      float m3 = start + step * (float)(4 * r4 + 3); float t3 = ed - m3;
      v.x = C * __expf(-beta * t0 * t0) * em;
      v.y = C * __expf(-beta * t1 * t1) * em;
      v.z = C * __expf(-beta * t2 * t2) * em;
      v.w = C * __expf(-beta * t3 * t3) * em;
    }
    eav[r4] = v;
  }

  float inv = (ew > 0.0f) ? (1.0f / ew) : 1.0f;
  float ux = rx * inv, uy = ry * inv, uz = rz * inv;

  int b  = sg / PAT;
  int r1 = (sg % PAT) / AAT;
  int r2 = (dg % PAT) / AAT;
  int keep = emi && smask[b * LRES + r1] && smask[b * LRES + r2];
  float kf = keep ? 1.0f : 0.0f;
  int mm = emi && (sg != dg);

  scaleW[e]   = C * (mm ? 1.0f : 0.0f);
  out_ew[e]   = ew * kf;
  out_keep[e] = kf;

  const float s3 = 1.7320508075688772f;
  float4 sh0, sh1;
  sh0.x = ux * kf; sh0.y = uy * kf; sh0.z = uz * kf;
  sh0.w = s3 * ux * uz * kf;
  sh1.x = s3 * ux * uy * kf;
  sh1.y = (uy * uy - 0.5f * (ux * ux + uz * uz)) * kf;
  sh1.z = s3 * uy * uz * kf;
  sh1.w = 0.5f * s3 * (uz * uz - ux * ux) * kf;
  float4* shv = (float4*)(out_sh + (size_t)e * 8);
  shv[0] = sh0; shv[1] = sh1;
}

// ---------------- kernel 4: f32 WMMA GEMM: D = (A[MxK] @ B[Kx128] + bias) * rowscale?
// One block = 16 rows x 128 cols; 8 waves, each wave owns one 16-col tile and
// computes its 16x16 D tile via V_WMMA_F32_16X16X4_F32, K stepped by 4.
// A-fragment layout (ISA 32-bit A 16x4): lanes 0-15 -> M=lane, regs K0,K1;
// lanes 16-31 -> same M, regs K2,K3. C/D: reg r -> rows r (lanes 0-15) and
// r+8 (lanes 16-31), col = lane&15.
__global__ __launch_bounds__(256) void k_gemm_wmma(const float* __restrict__ A, int K,
                                                   const float* __restrict__ B,
                                                   const float* __restrict__ bias,
                                                   const float* __restrict__ rowscale,
                                                   float* __restrict__ Dst, int M) {
  __shared__ float As[16 * 256];   // supports K up to 256
  const int rowbase = blockIdx.x * 16;
  for (int idx = threadIdx.x; idx < 16 * K; idx += 256) {
    As[idx] = A[(size_t)(rowbase + idx / K) * K + (idx % K)];
    // CDNA5 prefetch path (global_prefetch_b8): pull next block's A tile
    if (rowbase + 16 < M)
      __builtin_prefetch(&A[(size_t)(rowbase + 16 + idx / K) * K + (idx % K)], 0, 0);
  }
  __syncthreads();

  const int lane = threadIdx.x & 31;
  const int wave = threadIdx.x >> 5;
  const int lh   = lane >> 4;
  const int lm   = lane & 15;
  const int colg = wave * 16 + lm;

  v8f c = {};
  for (int k0 = 0; k0 < K; k0 += 4) {
    int ka = k0 + 2 * lh;
    v2f a, b;
    a[0] = As[lm * K + ka];
    a[1] = As[lm * K + ka + 1];
    b[0] = B[(size_t)ka * FEATD + colg];
    b[1] = B[(size_t)(ka + 1) * FEATD + colg];
    c = __builtin_amdgcn_wmma_f32_16x16x4_f32(false, a, false, b,
                                              (short)0, c, false, false);
  }

  float bs = bias[colg];
#pragma unroll
  for (int r = 0; r < 8; ++r) {
    int rowg = rowbase + r + 8 * lh;
    float v = c[r] + bs;
    if (rowscale) v *= rowscale[rowg];
    Dst[(size_t)rowg * FEATD + colg] = v;
  }
}

// ---------------- kernel 5: agg (no atomics: dst(e)=e/16) + concat ----
// One thread = one atom x 4 features (float4 / b128 path).
__global__ void k_agg(const int*   __restrict__ aaa,
                      const int*   __restrict__ srcIdx,
                      const float* __restrict__ Wbuf,   // E x 128
                      const float* __restrict__ emb,    // 17 x 128
                      const float* __restrict__ nemb,   // 17 x 128
                      float* __restrict__ concatAF) {   // NA x 256
  int i = blockIdx.x * blockDim.x + threadIdx.x;      // NA * 32
  if (i >= NATOT * (FEATD / 4)) return;
  int a  = i / (FEATD / 4);
  int f4 = (i % (FEATD / 4)) * 4;
  int av = aaa[a];
  float4 ft = make_float4(0.f, 0.f, 0.f, 0.f);
  if (av != 15) ft = *(const float4*)(emb + (size_t)av * FEATD + f4);
  float4 s = make_float4(0.f, 0.f, 0.f, 0.f);
#pragma unroll
  for (int k = 0; k < KNN; ++k) {
    int e  = a * KNN + k;
    int sg = srcIdx[e];
    float4 xn = *(const float4*)(nemb + (size_t)aaa[sg] * FEATD + f4);
    float4 wv = *(const float4*)(Wbuf + (size_t)e * FEATD + f4);
    s.x += xn.x * wv.x; s.y += xn.y * wv.y;
    s.z += xn.z * wv.z; s.w += xn.w * wv.w;
  }
  *(float4*)(concatAF + (size_t)a * (2 * FEATD) + f4)         = ft;
  *(float4*)(concatAF + (size_t)a * (2 * FEATD) + FEATD + f4) = s;
}

// ---------------- kernel 6: eout = (x[src]+x[dst]) * Be * keep --------
// One thread = one edge x 4 features (b128 loads/stores).
__global__ void k_eout(const float* __restrict__ x,
                       const int*   __restrict__ srcIdx,
                       const float* __restrict__ Be,     // E x 128 (edge_attr@We + be)
                       const float* __restrict__ keepf,  // E
                       float* __restrict__ eout) {
  size_t i = (size_t)blockIdx.x * blockDim.x + threadIdx.x;  // E * 32
  if (i >= (size_t)NEDGE * (FEATD / 4)) return;
  size_t e = i / (FEATD / 4);
  int   f4 = (int)(i % (FEATD / 4)) * 4;
  int sg = srcIdx[e];
  int dg = (int)(e / KNN);
  float kf = keepf[e];
  float4 xs = *(const float4*)(x + (size_t)sg * FEATD + f4);
  float4 xd = *(const float4*)(x + (size_t)dg * FEATD + f4);
  float4 bv = *(const float4*)(Be + (size_t)e * FEATD + f4);
  float4 o;
  o.x = (xs.x + xd.x) * bv.x * kf;
  o.y = (xs.y + xd.y) * bv.y * kf;
  o.z = (xs.z + xd.z) * bv.z * kf;
  o.w = (xs.w + xd.w) * bv.w * kf;
  *(float4*)(eout + (size_t)e * FEATD + f4) = o;
}

// ---------------- kernel 7: zero the "vec" output block (b128) -------
__global__ void k_zero(float4* __restrict__ p, unsigned long long n4) {
  size_t i = (size_t)blockIdx.x * blockDim.x + threadIdx.x;
  if (i < n4) p[i] = make_float4(0.f, 0.f, 0.f, 0.f);
}

// ---------------- launcher -------------------------------------------
extern "C" void kernel_launch(void* const* d_in, const int* in_sizes, int n_in,
                              void* d_out, int out_size, void* d_ws, size_t ws_size,
                              hipStream_t stream) {
  (void)in_sizes; (void)n_in; (void)out_size; (void)ws_size;
  // inputs (setup_inputs order)
  const float* pos        = (const float*)d_in[3];   // N,L,A,3
  const int*   mask_atoms = (const int*)  d_in[4];
  const int*   smask      = (const int*)  d_in[6];
  const int*   seqmask    = (const int*)  d_in[7];
  const float* emb        = (const float*)d_in[8];
  const float* nemb       = (const float*)d_in[9];
  const float* Wd         = (const float*)d_in[10];
  const float* bd         = (const float*)d_in[11];
  const float* Wc         = (const float*)d_in[12];
  const float* bc         = (const float*)d_in[13];
  const float* We         = (const float*)d_in[14];
  const float* be         = (const float*)d_in[15];

  // outputs, concatenated flat in return order
  float* out    = (float*)d_out;
  float* o_x    = out;                                       // NA*128
  float* o_vec  = o_x   + (size_t)NATOT * FEATD;             // NA*8*128 (zeros)
  int*   o_ei   = (int*)(o_vec + (size_t)NATOT * 8 * FEATD); // 2*E ints
  float* o_ew   = (float*)(o_ei + 2 * (size_t)NEDGE);        // E
  float* o_eout = o_ew  + (size_t)NEDGE;                     // E*128
  float* o_sh   = o_eout + (size_t)NEDGE * FEATD;            // E*8
  float* o_keep = o_sh  + (size_t)NEDGE * 8;                 // E

  // workspace carve (~176 MB): Be GEMM reuses Wbuf after agg consumes it
  char* w = (char*)d_ws;
  int*   aaa       = (int*)w;   w += sizeof(int)   * (size_t)NATOT;
  int*   srcIdx    = (int*)w;   w += sizeof(int)   * (size_t)NEDGE;
  int*   emaskE    = (int*)w;   w += sizeof(int)   * (size_t)NEDGE;
  float* edge_attr = (float*)w; w += sizeof(float) * (size_t)NEDGE * NRBF;
  float* scaleW    = (float*)w; w += sizeof(float) * (size_t)NEDGE;
  float* Wbuf      = (float*)w; w += sizeof(float) * (size_t)NEDGE * FEATD;
  float* concatAF  = (float*)w; w += sizeof(float) * (size_t)NATOT * 2 * FEATD;

  k_aaa<<<(NATOT + 255) / 256, 256, 0, stream>>>(mask_atoms, seqmask, aaa);

  dim3 gknn(PAT / 256, NBX);
  k_knn<<<gknn, 256, 0, stream>>>(pos, mask_atoms, srcIdx, emaskE, o_ei);

  k_edge<<<(NEDGE + 255) / 256, 256, 0, stream>>>(pos, srcIdx, emaskE, smask,
                                                  edge_attr, scaleW,
                                                  o_ew, o_sh, o_keep);

  // W = (edge_attr @ Wd + bd) * (C*mmask)   -> Wbuf
  k_gemm_wmma<<<NEDGE / 16, 256, 0, stream>>>(edge_attr, NRBF, Wd, bd, scaleW,
                                              Wbuf, NEDGE);

  k_agg<<<(NATOT * (FEATD / 4) + 255) / 256, 256, 0, stream>>>(aaa, srcIdx, Wbuf,
                                                               emb, nemb, concatAF);

  // Be = edge_attr @ We + be   -> reuse Wbuf (agg already consumed it)
  k_gemm_wmma<<<NEDGE / 16, 256, 0, stream>>>(edge_attr, NRBF, We, be, nullptr,
                                              Wbuf, NEDGE);

  // x = concat(feat, agg) @ Wc + bc  -> straight into output
  k_gemm_wmma<<<NATOT / 16, 256, 0, stream>>>(concatAF, 2 * FEATD, Wc, bc, nullptr,
                                              o_x, NATOT);

  size_t ne4 = (size_t)NEDGE * (FEATD / 4);
  k_eout<<<(unsigned)((ne4 + 255) / 256), 256, 0, stream>>>(o_x, srcIdx, Wbuf,
                                                            o_keep, o_eout);

  size_t nv4 = (size_t)NATOT * 8 * FEATD / 4;
  k_zero<<<(unsigned)((nv4 + 255) / 256), 256, 0, stream>>>((float4*)o_vec,
                                                            (unsigned long long)nv4);
}